// ReconstructGraph_42004780155134
// MI455X (gfx1250) — compile-verified
//
#include <hip/hip_runtime.h>
#include <math.h>

// Problem constants from the reference: B=32, N=512, D=64, H=32
#define BB 32
#define NN 512
#define DD 64
#define HH 32

typedef __attribute__((ext_vector_type(2))) float v2f;
typedef __attribute__((ext_vector_type(8))) float v8f;

// gfx1250 has V_TANH_F32 in the TRANS pipe; use the builtin when the
// toolchain declares it, else fall back to libm tanhf (ocml polynomial).
#if defined(__has_builtin)
#if __has_builtin(__builtin_amdgcn_tanhf)
#define TANH_F32(x) __builtin_amdgcn_tanhf(x)
#endif
#endif
#ifndef TANH_F32
#define TANH_F32(x) tanhf(x)
#endif

// ---------------------------------------------------------------------------
// Kernel 1: q = attendant @ w_q, k = receiver @ w_k   ([B*N, D] @ [D, H])
// One thread per (row, h); w matrices staged in LDS (8KB each).
// ---------------------------------------------------------------------------
__global__ __launch_bounds__(256) void proj_kernel(
    const float* __restrict__ receiver, const float* __restrict__ attendant,
    const float* __restrict__ w_q, const float* __restrict__ w_k,
    float* __restrict__ qout, float* __restrict__ kout)
{
    __shared__ float wq[DD * HH];
    __shared__ float wk[DD * HH];
    const int tid = threadIdx.x;
    for (int i = tid; i < DD * HH; i += 256) { wq[i] = w_q[i]; wk[i] = w_k[i]; }
    __syncthreads();

    const int h   = tid & (HH - 1);
    const int r   = tid >> 5;                  // 8 rows per block
    const int row = blockIdx.x * 8 + r;        // row in [0, B*N)

    const float4* arow = (const float4*)(attendant + (size_t)row * DD);
    const float4* rrow = (const float4*)(receiver  + (size_t)row * DD);

    float accq = 0.f, acck = 0.f;
#pragma unroll
    for (int d4 = 0; d4 < DD / 4; ++d4) {
        const float4 av = arow[d4];
        const float4 rv = rrow[d4];
        const int d = d4 * 4;
        accq = fmaf(av.x, wq[(d + 0) * HH + h], accq);
        accq = fmaf(av.y, wq[(d + 1) * HH + h], accq);
        accq = fmaf(av.z, wq[(d + 2) * HH + h], accq);
        accq = fmaf(av.w, wq[(d + 3) * HH + h], accq);
        acck = fmaf(rv.x, wk[(d + 0) * HH + h], acck);
        acck = fmaf(rv.y, wk[(d + 1) * HH + h], acck);
        acck = fmaf(rv.z, wk[(d + 2) * HH + h], acck);
        acck = fmaf(rv.w, wk[(d + 3) * HH + h], acck);
    }
    qout[(size_t)row * HH + h] = accq;
    kout[(size_t)row * HH + h] = acck;
}

// ---------------------------------------------------------------------------
// Kernel 2: scores[b,n,m] = sum_h a[h] * tanh(q[b,m,h] + k[b,n,h] + bias[h])
// 16x16 (n,m) tile per block; q/k rows staged in LDS with stride-33 padding.
// ---------------------------------------------------------------------------
__global__ __launch_bounds__(256) void scores_kernel(
    const float* __restrict__ q, const float* __restrict__ k,
    const float* __restrict__ bias, const float* __restrict__ a,
    float* __restrict__ scores)
{
    __shared__ float qs[16 * 33];   // q rows for m-tile, bias pre-added
    __shared__ float ks[16 * 33];   // k rows for n-tile
    __shared__ float av[HH];

    const int b  = blockIdx.z;
    const int m0 = blockIdx.x * 16;
    const int n0 = blockIdx.y * 16;
    const int tid = threadIdx.x;

    for (int i = tid; i < 16 * HH; i += 256) {
        const int rr = i >> 5;
        const int hh = i & (HH - 1);
        qs[rr * 33 + hh] = q[((size_t)b * NN + m0 + rr) * HH + hh] + bias[hh];
        ks[rr * 33 + hh] = k[((size_t)b * NN + n0 + rr) * HH + hh];
    }
    if (tid < HH) av[tid] = a[tid];
    __syncthreads();

    const int lm = tid & 15;
    const int ln = tid >> 4;
    float s = 0.f;
#pragma unroll
    for (int h = 0; h < HH; ++h) {
        const float x = qs[lm * 33 + h] + ks[ln * 33 + h];
        s = fmaf(av[h], TANH_F32(x), s);
    }
    scores[(size_t)b * NN * NN + (size_t)(n0 + ln) * NN + (m0 + lm)] = s;
}

// ---------------------------------------------------------------------------
// Kernel 3 (used twice): batched f32 GEMM C[b] = A[b] @ Bm[b], all [N,N]
// row-major. Wave32 tile = 32x32 via 4x V_WMMA_F32_16X16X4_F32 accumulators.
// Fragment layouts per CDNA5 ISA 7.12.2:
//   A 16x4 : lanes 0-15 -> M=lane, K=0,1 in v0,v1 ; lanes 16-31 -> K=2,3
//   B 4x16 : lanes 0-15 -> N=lane, K=0,1 in v0,v1 ; lanes 16-31 -> K=2,3
//   C/D    : vgpr v, lanes 0-15 -> (M=v,   N=lane) ; lanes 16-31 -> (M=v+8)
// ---------------------------------------------------------------------------
__global__ __launch_bounds__(256) void gemm_kernel(
    const float* __restrict__ A, const float* __restrict__ Bm,
    float* __restrict__ C)
{
    const int b = blockIdx.z;
    const float* __restrict__ Ab = A  + (size_t)b * NN * NN;
    const float* __restrict__ Bb = Bm + (size_t)b * NN * NN;
    float* __restrict__ Cb       = C  + (size_t)b * NN * NN;

    const int tid  = threadIdx.x;
    const int lane = tid & 31;
    const int wave = tid >> 5;        // 8 waves
    const int wm   = wave >> 2;       // 0..1
    const int wn   = wave & 3;        // 0..3
    const int M0   = blockIdx.y * 64  + wm * 32;
    const int N0   = blockIdx.x * 128 + wn * 32;

    const bool hi = (lane >= 16);
    const int  l  = lane & 15;
    const int  ksel = hi ? 2 : 0;     // this half-wave owns K=2,3 (else K=0,1)

    v8f acc00 = {0.f,0.f,0.f,0.f,0.f,0.f,0.f,0.f};
    v8f acc01 = acc00, acc10 = acc00, acc11 = acc00;

    for (int kk = 0; kk < NN; kk += 4) {
        const int ak = kk + ksel;
        // A fragments: two consecutive f32 along K of rows M0+l / M0+16+l
        const v2f afr0 = *(const v2f*)(Ab + (size_t)(M0 +      l) * NN + ak);
        const v2f afr1 = *(const v2f*)(Ab + (size_t)(M0 + 16 + l) * NN + ak);
        // B fragments: column N0+l (and +16), rows ak and ak+1
        const float* bp0 = Bb + (size_t)ak * NN + N0 + l;
        const float* bp1 = bp0 + 16;
        const v2f bfr0 = { bp0[0], bp0[NN] };
        const v2f bfr1 = { bp1[0], bp1[NN] };

        acc00 = __builtin_amdgcn_wmma_f32_16x16x4_f32(false, afr0, false, bfr0,
                                                      (short)0, acc00, false, false);
        acc01 = __builtin_amdgcn_wmma_f32_16x16x4_f32(false, afr0, false, bfr1,
                                                      (short)0, acc01, false, false);
        acc10 = __builtin_amdgcn_wmma_f32_16x16x4_f32(false, afr1, false, bfr0,
                                                      (short)0, acc10, false, false);
        acc11 = __builtin_amdgcn_wmma_f32_16x16x4_f32(false, afr1, false, bfr1,
                                                      (short)0, acc11, false, false);
    }

#pragma unroll
    for (int v = 0; v < 8; ++v) {
        const int ro = v + (hi ? 8 : 0);
        Cb[(size_t)(M0 +      ro) * NN + N0 +      l] = acc00[v];
        Cb[(size_t)(M0 +      ro) * NN + N0 + 16 + l] = acc01[v];
        Cb[(size_t)(M0 + 16 + ro) * NN + N0 +      l] = acc10[v];
        Cb[(size_t)(M0 + 16 + ro) * NN + N0 + 16 + l] = acc11[v];
    }
}

// ---------------------------------------------------------------------------
// Launch: proj -> scores -> t = adj@scores -> out = t@adj
// Workspace layout (floats): q[B*N*H] | k[B*N*H] | scores[B*N*N] | t[B*N*N]
// total = 2*0.5M + 2*8M floats ~= 68 MB
// ---------------------------------------------------------------------------
extern "C" void kernel_launch(void* const* d_in, const int* in_sizes, int n_in,
                              void* d_out, int out_size, void* d_ws, size_t ws_size,
                              hipStream_t stream) {
    const float* receiver  = (const float*)d_in[0];
    const float* attendant = (const float*)d_in[1];
    const float* adj       = (const float*)d_in[2];
    const float* w_q       = (const float*)d_in[3];
    const float* w_k       = (const float*)d_in[4];
    const float* bias      = (const float*)d_in[5];
    const float* avec      = (const float*)d_in[6];
    float* out = (float*)d_out;

    float* q      = (float*)d_ws;
    float* k      = q + (size_t)BB * NN * HH;
    float* scores = k + (size_t)BB * NN * HH;
    float* t      = scores + (size_t)BB * NN * NN;

    proj_kernel<<<(BB * NN) / 8, 256, 0, stream>>>(receiver, attendant,
                                                   w_q, w_k, q, k);
    scores_kernel<<<dim3(NN / 16, NN / 16, BB), 256, 0, stream>>>(q, k, bias,
                                                                  avec, scores);
    gemm_kernel<<<dim3(NN / 128, NN / 64, BB), 256, 0, stream>>>(adj, scores, t);
    gemm_kernel<<<dim3(NN / 128, NN / 64, BB), 256, 0, stream>>>(t, adj, out);
}